// TtLlamaAttention_optimized_5119601017509
// MI455X (gfx1250) — compile-verified
//
#include <hip/hip_runtime.h>
#include <math.h>

typedef float v2f __attribute__((ext_vector_type(2)));
typedef float v4f __attribute__((ext_vector_type(4)));
typedef float v8f __attribute__((ext_vector_type(8)));
typedef unsigned int v4u __attribute__((ext_vector_type(4)));
typedef int v4i_t __attribute__((ext_vector_type(4)));
typedef int v8i_t __attribute__((ext_vector_type(8)));

#define WMMA_F32(a, b, c) \
  __builtin_amdgcn_wmma_f32_16x16x4_f32(false, (a), false, (b), (short)0, (c), false, false)

constexpr int kDim    = 8192;
constexpr int kNHeads = 64;
constexpr int kNKV    = 8;
constexpr int kHD     = 128;
constexpr int kBatch  = 32;
constexpr int kMaxSeq = 4096;
constexpr int kPadLen = 2048;   // nearest_32(START_POS + 1), START_POS = 2047

// ---------------------------------------------------------------------------
// Tensor Data Mover: async 2D tile load Global -> LDS (CDNA5 TDM, ISA ch.8).
// D# group0: count=1, lds byte addr, 57-bit global tile addr, type=2.
// D# group1: data_size=4B, LDS padding (to hit bank-conflict-free strides),
//            tensor dims, tile dims, dim0 stride. Groups 2/3 unused (2D).
// ---------------------------------------------------------------------------
__device__ __forceinline__ void tdm_load_2d(unsigned lds_off,        // LDS byte offset
                                            const void* gptr,       // tile start
                                            unsigned tensor_w,      // dim0 (elems)
                                            unsigned tensor_h,      // dim1 (rows)
                                            unsigned tile_w,        // tile dim0
                                            unsigned tile_h,        // tile dim1
                                            unsigned stride_elems,  // dim0 stride
                                            unsigned pad_interval,  // 5 => every 64 dw
                                            unsigned pad_amount) {  // n => n+1 pad dw
  unsigned long long ga = (unsigned long long)(uintptr_t)gptr;
  v4u g0;
  g0.x = 1u;                                            // count=1, user mode
  g0.y = lds_off;                                       // lds_addr
  g0.z = (unsigned)(ga & 0xffffffffu);                  // global_addr[31:0]
  g0.w = (unsigned)((ga >> 32) & 0x1ffffffu) | (2u << 30);  // [56:32] | type=2
  v8i_t g1;
  g1[0] = (int)((2u << 16) | (1u << 20) | (pad_interval << 22) | (pad_amount << 25));
  g1[1] = (int)((tensor_w & 0xffffu) << 16);            // tensor_dim0[15:0]
  g1[2] = (int)((tensor_w >> 16) | ((tensor_h & 0xffffu) << 16));
  g1[3] = (int)((tensor_h >> 16) | (tile_w << 16));     // tile_dim0
  g1[4] = (int)tile_h;                                  // tile_dim1, tile_dim2=0
  g1[5] = (int)stride_elems;                            // tensor_dim0_stride[31:0]
  g1[6] = 0;                                            // stride hi, dim1_stride lo
  g1[7] = 0;
  v4i_t gz = {0, 0, 0, 0};
#if defined(__clang_major__) && (__clang_major__ >= 23)
  v8i_t gz8 = {0, 0, 0, 0, 0, 0, 0, 0};
  __builtin_amdgcn_tensor_load_to_lds(g0, g1, gz, gz, gz8, 0);
#else
  __builtin_amdgcn_tensor_load_to_lds(g0, g1, gz, gz, 0);
#endif
}

__device__ __forceinline__ unsigned lds_offset_of(const void* p) {
  // addrspacecast(LDS->flat) puts the LDS byte offset in the low 32 bits.
  return (unsigned)(unsigned long long)(uintptr_t)p;
}

// ---------------------------------------------------------------------------
// C(32 x N) = A(32 x K) @ B(K x N), row-major fp32, WMMA 16x16x4.
// TDM double-buffered staging: wave 0 issues A/B tile DMAs one chunk ahead,
// waits TENSORcnt<=2 (in-order completion => previous chunk landed), barrier
// publishes LDS, 8 waves each run 16 WMMAs per 64-K chunk.
// ---------------------------------------------------------------------------
__global__ __launch_bounds__(256) void gemm32(const float* __restrict__ A,
                                              const float* __restrict__ B,
                                              float* __restrict__ C,
                                              int K, int N) {
  __shared__ float As[2][32][68];   // 64 data dw + 4 pad dw per row (TDM pad)
  __shared__ float Bs[2][64][72];   // 64 data dw + 8 pad dw per row (TDM pad)

  const int tid  = threadIdx.x;
  const int lane = tid & 31;
  const int wave = tid >> 5;      // 0..7
  const int mt   = wave & 1;      // M tile: rows mt*16
  const int nt   = wave >> 1;     // N tile: cols nt*16
  const int half = lane >> 4;     // 0 | 1 -> K pair within fragment
  const int l16  = lane & 15;
  const int n0   = blockIdx.x * 64;
  const int nchunks = K / 64;

  const unsigned a_off[2] = {lds_offset_of(&As[0][0][0]), lds_offset_of(&As[1][0][0])};
  const unsigned b_off[2] = {lds_offset_of(&Bs[0][0][0]), lds_offset_of(&Bs[1][0][0])};

  if (wave == 0) {  // uniform branch: one wave owns the TDM queue
    tdm_load_2d(a_off[0], A, K, 32, 64, 32, K, 5u, 3u);
    tdm_load_2d(b_off[0], B + n0, N, K, 64, 64, N, 5u, 7u);
  }

  v8f acc = {};
  for (int c = 0; c < nchunks; ++c) {
    if (wave == 0) {
      if (c + 1 < nchunks) {
        const int k1 = (c + 1) * 64;
        tdm_load_2d(a_off[(c + 1) & 1], A + k1, K, 32, 64, 32, K, 5u, 3u);
        tdm_load_2d(b_off[(c + 1) & 1], B + (size_t)k1 * N + n0, N, K, 64, 64, N, 5u, 7u);
        __builtin_amdgcn_s_wait_tensorcnt(2);   // chunk c complete
      } else {
        __builtin_amdgcn_s_wait_tensorcnt(0);
      }
      asm volatile("" ::: "memory");  // TDM wrote LDS: block ds-load folding
    }
    __syncthreads();

    const float(*Asb)[68] = As[c & 1];
    const float(*Bsb)[72] = Bs[c & 1];
#pragma unroll
    for (int kk = 0; kk < 64; kk += 4) {
      // A frag: lane = row m (l16), K = kk + half*2 .. +1
      v2f a = *(const v2f*)&Asb[mt * 16 + l16][kk + half * 2];
      // B frag: VGPR0 = row kk+half*2, VGPR1 = +1, col = l16
      v2f b;
      b.x = Bsb[kk + half * 2 + 0][nt * 16 + l16];
      b.y = Bsb[kk + half * 2 + 1][nt * 16 + l16];
      acc = WMMA_F32(a, b, acc);
    }
    __syncthreads();
  }

  // C frag: VGPR r -> M = mt*16 + half*8 + r, N = n0 + nt*16 + l16
  const int m = mt * 16 + half * 8;
  const int n = n0 + nt * 16 + l16;
#pragma unroll
  for (int r = 0; r < 8; ++r) {
    C[(size_t)(m + r) * N + n] = acc[r];
  }
}

// ---------------------------------------------------------------------------
// RoPE: rot_mat is the standard 2x2 block rotation; read cos/sin from its
// diagonal and rotate pairs in place (ws buffer only, never an input).
// ---------------------------------------------------------------------------
__global__ void rope(float* __restrict__ q, const float* __restrict__ rot, int nrows) {
  int idx = blockIdx.x * blockDim.x + threadIdx.x;   // one thread per pair
  if (idx >= nrows * (kHD / 2)) return;
  int row = idx >> 6;
  int i   = idx & 63;
  float c = rot[(2 * i) * kHD + 2 * i];       // cos
  float s = rot[(2 * i) * kHD + 2 * i + 1];   // sin
  float x0 = q[row * kHD + 2 * i];
  float x1 = q[row * kHD + 2 * i + 1];
  q[row * kHD + 2 * i]     = x0 * c - x1 * s;
  q[row * kHD + 2 * i + 1] = x0 * s + x1 * c;
}

// ---------------------------------------------------------------------------
// Attention: one block per (batch, kv_head). 8 q-heads padded to M=16.
// Scores via WMMA (K=128), softmax in LDS, P@V via WMMA (K=2048).
// The start_pos row of the cache is substituted with the fresh k/v at read
// time (inputs are never mutated).
// ---------------------------------------------------------------------------
__global__ __launch_bounds__(256) void attn(const float* __restrict__ qr,      // 32 x 64 x 128
                                            const float* __restrict__ kn,      // 32 x 8 x 128
                                            const float* __restrict__ vn,      // 32 x 8 x 128
                                            const float* __restrict__ cache_k, // 32 x 8 x 4096 x 128
                                            const float* __restrict__ cache_v,
                                            float* __restrict__ out,           // 32 x 64 x 128
                                            const int* __restrict__ sp) {
  __shared__ float S[8][kPadLen + 4];   // probs, padded stride (bank spread)
  __shared__ float row_sum[8];

  const int b    = blockIdx.x >> 3;
  const int g    = blockIdx.x & 7;
  const int tid  = threadIdx.x;
  const int lane = tid & 31;
  const int wave = tid >> 5;
  const int half = lane >> 4;
  const int l16  = lane & 15;
  const int start_pos = sp[0];
  const float scale = 0.08838834764831845f;   // 1/sqrt(128)

  // ---- Preload Q fragments (16x128 A-matrix, rows 8..15 zero), scale folded in.
  const float* qbase = qr + ((size_t)b * kNHeads + g * 8) * kHD;
  v2f qf[32];
#pragma unroll
  for (int i = 0; i < 32; ++i) {
    int k0 = i * 4 + half * 2;
    if (l16 < 8) {
      v2f t = *(const v2f*)(qbase + l16 * kHD + k0);
      qf[i].x = t.x * scale;
      qf[i].y = t.y * scale;
    } else {
      qf[i].x = 0.0f;
      qf[i].y = 0.0f;
    }
  }

  // ---- Phase 1: scores S[h][s] = (q_h . k_s) * scale, masked.
  const float* kc  = cache_k + ((size_t)b * kNKV + g) * kMaxSeq * kHD;
  const float* knb = kn + ((size_t)b * kNKV + g) * kHD;

  for (int t = wave; t < kPadLen / 16; t += 8) {
    const int s = t * 16 + l16;                      // B column -> seq index
    const float* kp = (s == start_pos) ? knb : (kc + (size_t)s * kHD);
    v8f acc = {};
#pragma unroll
    for (int i = 0; i < 32; ++i) {
      int k0 = i * 4 + half * 2;
      v2f bf = *(const v2f*)(kp + k0);               // {K[s][k0], K[s][k0+1]}
      acc = WMMA_F32(qf[i], bf, acc);
    }
    if (half == 0) {                                 // rows 0..7 = valid heads
      const bool valid = (s <= start_pos);
#pragma unroll
      for (int r = 0; r < 8; ++r) {
        S[r][t * 16 + l16] = valid ? acc[r] : -1e30f;
      }
    }
  }
  __syncthreads();

  // ---- Phase 2: softmax (unnormalized exp in LDS, sums kept aside).
  {
    const int h = wave;                              // 8 waves -> 8 heads
    float mx = -INFINITY;
    for (int i = lane; i < kPadLen; i += 32) mx = fmaxf(mx, S[h][i]);
#pragma unroll
    for (int off = 16; off > 0; off >>= 1) mx = fmaxf(mx, __shfl_xor(mx, off, 32));
    float sum = 0.0f;
    for (int i = lane; i < kPadLen; i += 32) {
      float e = __expf(S[h][i] - mx);
      S[h][i] = e;
      sum += e;
    }
#pragma unroll
    for (int off = 16; off > 0; off >>= 1) sum += __shfl_xor(sum, off, 32);
    if (lane == 0) row_sum[h] = sum;
  }
  __syncthreads();

  // ---- Phase 3: attn = P @ V, wave w owns output cols [w*16, w*16+16).
  const float* vc  = cache_v + ((size_t)b * kNKV + g) * kMaxSeq * kHD;
  const float* vnb = vn + ((size_t)b * kNKV + g) * kHD;
  const int n0 = wave * 16;
  v8f acc = {};
#pragma unroll 4
  for (int k0 = 0; k0 < kPadLen; k0 += 4) {
    v2f a;
    if (l16 < 8) {
      a = *(const v2f*)&S[l16][k0 + half * 2];       // P rows 0..7
    } else {
      a.x = 0.0f; a.y = 0.0f;                        // pad rows
    }
    const int s0 = k0 + half * 2;
    const float* vp0 = (s0     == start_pos) ? vnb : (vc + (size_t)s0 * kHD);
    const float* vp1 = (s0 + 1 == start_pos) ? vnb : (vc + (size_t)(s0 + 1) * kHD);
    v2f bf;
    bf.x = vp0[n0 + l16];
    bf.y = vp1[n0 + l16];
    acc = WMMA_F32(a, bf, acc);
  }
  if (half == 0) {
#pragma unroll
    for (int r = 0; r < 8; ++r) {
      out[((size_t)b * kNHeads + g * 8 + r) * kHD + n0 + l16] = acc[r] / row_sum[r];
    }
  }
}

// ---------------------------------------------------------------------------
// Host-side orchestration (all on `stream`; ws layout below).
// Inputs: 0:x 1:wq 2:wk 3:wv 4:wo 5:rot_mat 6:cache_k 7:cache_v 8:start_pos
// ---------------------------------------------------------------------------
extern "C" void kernel_launch(void* const* d_in, const int* in_sizes, int n_in,
                              void* d_out, int out_size, void* d_ws, size_t ws_size,
                              hipStream_t stream) {
  (void)in_sizes; (void)n_in; (void)out_size; (void)ws_size;

  const float* x       = (const float*)d_in[0];
  const float* wq      = (const float*)d_in[1];
  const float* wk      = (const float*)d_in[2];
  const float* wv      = (const float*)d_in[3];
  const float* wo      = (const float*)d_in[4];
  const float* rot     = (const float*)d_in[5];
  const float* cache_k = (const float*)d_in[6];
  const float* cache_v = (const float*)d_in[7];
  const int*   sp      = (const int*)d_in[8];
  float* out = (float*)d_out;

  float* qb = (float*)d_ws;                       // 32 x 8192
  float* kb = qb + (size_t)kBatch * kDim;         // 32 x 1024
  float* vb = kb + (size_t)kBatch * kNKV * kHD;   // 32 x 1024
  float* ab = vb + (size_t)kBatch * kNKV * kHD;   // 32 x 8192

  // QKV projections (WMMA f32 GEMMs, TDM-staged).
  gemm32<<<kDim / 64, 256, 0, stream>>>(x, wq, qb, kDim, kDim);
  gemm32<<<(kNKV * kHD) / 64, 256, 0, stream>>>(x, wk, kb, kDim, kNKV * kHD);
  gemm32<<<(kNKV * kHD) / 64, 256, 0, stream>>>(x, wv, vb, kDim, kNKV * kHD);

  // RoPE on q and k.
  {
    int pairs_q = kBatch * kNHeads * (kHD / 2);
    int pairs_k = kBatch * kNKV * (kHD / 2);
    rope<<<(pairs_q + 255) / 256, 256, 0, stream>>>(qb, rot, kBatch * kNHeads);
    rope<<<(pairs_k + 255) / 256, 256, 0, stream>>>(kb, rot, kBatch * kNKV);
  }

  // Attention: one block per (batch, kv_head).
  attn<<<kBatch * kNKV, 256, 0, stream>>>(qb, kb, vb, cache_k, cache_v, ab, sp);

  // Output projection.
  gemm32<<<kDim / 64, 256, 0, stream>>>(ab, wo, out, kDim, kDim);
}